// _HashGrid_18459769438224
// MI455X (gfx1250) — compile-verified
//
#include <hip/hip_runtime.h>
#include <stdint.h>

// ---------------------------------------------------------------------------
// Instant-NGP style hash-grid encode, 3D, 2 features, 2^19-entry table.
// Memory-bound: ~80 MB HBM traffic (x in, out), gathers hit L2 (table = 4 MB).
// CDNA5 path: TDM (tensor_load_to_lds) double-buffered staging of the point
// coordinates into LDS, synchronized with s_wait_tensorcnt.
// ---------------------------------------------------------------------------

#define BLOCK       256
#define TILE        256                 // points per tile
#define TILE_FLOATS (TILE * 3)          // 768 dwords = 3072 bytes per tile

#define HASH_MASK 0x7FFFFu              // HASHMAP_SIZE = 2^19
#define PRIME1    2654435761u
#define PRIME2    805459861u
#define RES       128.0f

typedef uint32_t u32;
typedef u32 __attribute__((ext_vector_type(4))) v4u;
typedef int __attribute__((ext_vector_type(4))) v4i;
typedef int __attribute__((ext_vector_type(8))) v8i;

#if __has_builtin(__builtin_amdgcn_tensor_load_to_lds) && \
    __has_builtin(__builtin_amdgcn_s_wait_tensorcnt)
#define USE_TDM 1
#else
#define USE_TDM 0
#endif

#if USE_TDM
// Build a D# for a contiguous 3072-byte tile copy (2D tensor, one row of 768
// dwords) and issue TENSOR_LOAD_TO_LDS. Issued by one lane; TDM ignores EXEC.
// This toolchain's builtin takes 6 operands (g0, g1, g2, g3, extra, cpol);
// groups 2/3 + extra are for >=3D tensors and are zero for this 2D copy.
__device__ __forceinline__ void tdm_stage_tile(const float* gsrc, u32 lds_byte) {
  const uint64_t ga = (uint64_t)(uintptr_t)gsrc;
  v4u g0;
  g0.x = 1u;                                    // count=1 (valid), user mode
  g0.y = lds_byte;                              // lds_addr [63:32]
  g0.z = (u32)ga;                               // global_addr [95:64]
  g0.w = (u32)((ga >> 32) & 0x01FFFFFFu)        // global_addr [120:96]
         | 0x80000000u;                         // type=2 ("image") [127:126]
  v8i g1;
  g1[0] = 0x00020000;                           // data_size=2 (4B), mask=0
  g1[1] = (int)((u32)TILE_FLOATS << 16);        // tensor_dim0 = 768 (bits 79:48)
  g1[2] = 0x00010000;                           // tensor_dim1 = 1   (bits 111:80)
  g1[3] = (int)((u32)TILE_FLOATS << 16);        // tile_dim0   = 768 (bits 127:112)
  g1[4] = 1;                                    // tile_dim1=1, tile_dim2=0
  g1[5] = TILE_FLOATS;                          // tensor_dim0_stride = 768
  g1[6] = 0;
  g1[7] = 0;
  v4i z4 = {0, 0, 0, 0};
  v8i z8 = {0, 0, 0, 0, 0, 0, 0, 0};
  __builtin_amdgcn_tensor_load_to_lds(g0, g1, z4, z4, z8, 0);
}
#endif

// Per-point trilinear hash-grid lookup. PRIME0==1 and the +1 corner hashes
// are hx+{0,P}: only two u32 multiplies per point; mod 2^19 is an AND.
__device__ __forceinline__ void compute_point(float X, float Y, float Z,
                                              const float2* __restrict__ emb2,
                                              float2* __restrict__ outp) {
  const float xs0 = X * RES, xs1 = Y * RES, xs2 = Z * RES;
  const float fl0 = floorf(xs0), fl1 = floorf(xs1), fl2 = floorf(xs2);
  const int   i0 = (int)fl0, i1 = (int)fl1, i2 = (int)fl2;
  const float t0 = xs0 - fl0, t1 = xs1 - fl1, t2 = xs2 - fl2;
  const float s0 = 1.0f - t0, s1 = 1.0f - t1, s2 = 1.0f - t2;

  const u32 hx0 = (u32)i0,          hx1 = hx0 + 1u;
  const u32 hy0 = (u32)i1 * PRIME1, hy1 = hy0 + PRIME1;
  const u32 hz0 = (u32)i2 * PRIME2, hz1 = hz0 + PRIME2;

  const u32 h0 = (hx0 ^ hy0 ^ hz0) & HASH_MASK;
  const u32 h1 = (hx1 ^ hy0 ^ hz0) & HASH_MASK;
  const u32 h2 = (hx0 ^ hy1 ^ hz0) & HASH_MASK;
  const u32 h3 = (hx1 ^ hy1 ^ hz0) & HASH_MASK;
  const u32 h4 = (hx0 ^ hy0 ^ hz1) & HASH_MASK;
  const u32 h5 = (hx1 ^ hy0 ^ hz1) & HASH_MASK;
  const u32 h6 = (hx0 ^ hy1 ^ hz1) & HASH_MASK;
  const u32 h7 = (hx1 ^ hy1 ^ hz1) & HASH_MASK;

  // Issue all 8 gathers (global_load_b64) before any use: full MLP overlap.
  const float2 e0 = emb2[h0], e1 = emb2[h1], e2 = emb2[h2], e3 = emb2[h3];
  const float2 e4 = emb2[h4], e5 = emb2[h5], e6 = emb2[h6], e7 = emb2[h7];

  const float w0 = s0 * s1 * s2;
  const float w1 = t0 * s1 * s2;
  const float w2 = s0 * t1 * s2;
  const float w3 = t0 * t1 * s2;
  const float w4 = s0 * s1 * t2;
  const float w5 = t0 * s1 * t2;
  const float w6 = s0 * t1 * t2;
  const float w7 = t0 * t1 * t2;

  float a0 = e0.x * w0,        a1 = e0.y * w0;
  a0 = fmaf(e1.x, w1, a0);     a1 = fmaf(e1.y, w1, a1);
  a0 = fmaf(e2.x, w2, a0);     a1 = fmaf(e2.y, w2, a1);
  a0 = fmaf(e3.x, w3, a0);     a1 = fmaf(e3.y, w3, a1);
  a0 = fmaf(e4.x, w4, a0);     a1 = fmaf(e4.y, w4, a1);
  a0 = fmaf(e5.x, w5, a0);     a1 = fmaf(e5.y, w5, a1);
  a0 = fmaf(e6.x, w6, a0);     a1 = fmaf(e6.y, w6, a1);
  a0 = fmaf(e7.x, w7, a0);     a1 = fmaf(e7.y, w7, a1);

  *outp = make_float2(a0, a1);   // global_store_b64
}

__global__ __launch_bounds__(BLOCK) void hashgrid_fwd(
    const float* __restrict__ x,
    const float* __restrict__ emb,
    float* __restrict__ out,
    int npoints, int ntiles) {
  const float2* emb2 = (const float2*)emb;
  float2* out2 = (float2*)out;
  const int tid = threadIdx.x;

#if USE_TDM
  __shared__ float sx[2][TILE_FLOATS];
  // Flat pointer to LDS: low 32 bits are the workgroup-relative LDS byte addr.
  const u32 lds_base = (u32)(uintptr_t)&sx[0][0];

  int tile = blockIdx.x;
  if (tile < ntiles) {
    if (tid == 0) tdm_stage_tile(x + (size_t)tile * TILE_FLOATS, lds_base);
    int parity = 0;
    for (; tile < ntiles; tile += (int)gridDim.x) {
      const int next = tile + (int)gridDim.x;
      if (tid == 0) {
        if (next < ntiles) {
          tdm_stage_tile(x + (size_t)next * TILE_FLOATS,
                         lds_base + (u32)((parity ^ 1) * (TILE_FLOATS * 4)));
          __builtin_amdgcn_s_wait_tensorcnt(1);   // current tile landed
        } else {
          __builtin_amdgcn_s_wait_tensorcnt(0);   // drain
        }
      }
      __syncthreads();                            // staged data visible to all
      const float* sp = &sx[parity][tid * 3];
      const int p = tile * TILE + tid;
      compute_point(sp[0], sp[1], sp[2], emb2, &out2[p]);
      __syncthreads();                            // done reading before reuse
      parity ^= 1;
    }
  }
#else
  for (int tile = blockIdx.x; tile < ntiles; tile += (int)gridDim.x) {
    const int next = tile + (int)gridDim.x;
    if (next < ntiles)
      __builtin_prefetch(x + (size_t)next * TILE_FLOATS + tid * 3, 0, 0);
    const int p = tile * TILE + tid;
    const float* xp = x + (size_t)p * 3;
    compute_point(xp[0], xp[1], xp[2], emb2, &out2[p]);
  }
#endif

  // Tail (npoints not a multiple of TILE; empty for N=4M).
  const int tail = ntiles * TILE;
  if (blockIdx.x == 0) {
    const int p = tail + tid;
    if (p < npoints) {
      const float* xp = x + (size_t)p * 3;
      compute_point(xp[0], xp[1], xp[2], emb2, &out2[p]);
    }
  }
}

extern "C" void kernel_launch(void* const* d_in, const int* in_sizes, int n_in,
                              void* d_out, int out_size, void* d_ws, size_t ws_size,
                              hipStream_t stream) {
  (void)n_in; (void)out_size; (void)d_ws; (void)ws_size;
  const float* x   = (const float*)d_in[0];
  const float* emb = (const float*)d_in[1];
  float* out       = (float*)d_out;

  const int npoints = in_sizes[0] / 3;
  const int ntiles  = npoints / TILE;   // full tiles; tail handled in-kernel

  int grid = ntiles > 0 ? ntiles : 1;
  if (grid > 2048) grid = 2048;         // ~7.6 tiles/block -> pipeline depth
  hashgrid_fwd<<<grid, BLOCK, 0, stream>>>(x, emb, out, npoints, ntiles);
}